// TEGTSNet_66331474919466
// MI455X (gfx1250) — compile-verified
//
#include <hip/hip_runtime.h>
#include <hip/hip_bf16.h>

// ---------------------------------------------------------------------------
// Problem constants (from reference)
// ---------------------------------------------------------------------------
constexpr int Bn = 16;    // batch
constexpr int Tn = 12;    // time
constexpr int Nn = 1024;  // nodes
constexpr int Dn = 64;    // in channels
constexpr int Fn = 64;    // conv filters
constexpr int CE = 192;   // (ORDER+1)*D
constexpr int COLS = Tn * Dn;  // 768 columns for graph GEMMs
constexpr float EPS = 1e-5f;

typedef float v2f __attribute__((ext_vector_type(2)));
typedef float v8f __attribute__((ext_vector_type(8)));

// fp32 matrix pipe: D(16x16) = A(16x4) * B(4x16) + C
__device__ __forceinline__ v8f wmma4(v2f a, v2f b, v8f c) {
  return __builtin_amdgcn_wmma_f32_16x16x4_f32(
      /*neg_a=*/false, a, /*neg_b=*/false, b,
      /*c_mod=*/(short)0, c, /*reuse_a=*/false, /*reuse_b=*/false);
}

// ---------------------------------------------------------------------------
// Kernel 0: zero the GRU hidden state
// ---------------------------------------------------------------------------
__global__ void zero_kernel(float* __restrict__ p, long long n) {
  long long i = (long long)blockIdx.x * blockDim.x + threadIdx.x;
  long long stride = (long long)gridDim.x * blockDim.x;
  for (; i < n; i += stride) p[i] = 0.0f;
}

// ---------------------------------------------------------------------------
// Kernel 1: batched graph GEMM  Out[b][w][j] = sum_v A[b][w][v]*Bsrc[v][j]
//   Bsrc element address = Bsrc + b*bStrideB + t*bStrideT + v*ldb + c  (j=t*64+c)
//   grid = (N/64, COLS/128, B), block = 128 (4 waves)
//   64x128 output tile per block; each wave: 16 rows x 8 col-tiles.
//   B panel staged pair-interleaved so each B-fragment is one ds_load_b64.
// ---------------------------------------------------------------------------
constexpr int XPITCH = 288;  // floats per kpair row: 288 % 64 == 32 -> half-wave
                             // bank sets disjoint for b64 fragment reads

__global__ __launch_bounds__(128) void graph_gemm_kernel(
    const float* __restrict__ Agraph,  // [B][N][N]
    const float* __restrict__ Bsrc,
    float* __restrict__ Out,           // [B][N][768]
    long long bStrideB, long long bStrideT, int ldb) {
  const int b    = blockIdx.z;
  const int tblk = blockIdx.y;         // 128-column block = 2 t-segments
  const int mblk = blockIdx.x;
  const int tid  = threadIdx.x;
  const int lane = tid & 31;
  const int wave = tid >> 5;
  const int lm   = lane & 15;
  const int kh   = (lane >> 4) * 2;    // 0 or 2 (A/B fragment K split)
  const int hi   = (lane >> 4) * 8;    // 0 or 8 (C/D row split)

  __shared__ float As[64][20];           // pitch 20: conflict-free b64 A-frags
  __shared__ float Xs2[8 * XPITCH];      // 8 kpairs x 128 cols, pair-interleaved

  const float* Ab = Agraph + (size_t)b * Nn * Nn + (size_t)mblk * 64 * Nn;
  const float* Bb = Bsrc + (size_t)b * bStrideB + (size_t)(2 * tblk) * bStrideT;
  float*       Ob = Out  + (size_t)b * Nn * COLS + (size_t)tblk * 128;

  v8f acc[8] = {};
  const int r0 = wave * 16;

  for (int k0 = 0; k0 < Nn; k0 += 16) {
    __syncthreads();
    {  // stage A panel: 64 rows x 16 K
      int r  = tid >> 2;
      int c4 = (tid & 3) * 4;
      float4 a0 = *(const float4*)(Ab + (size_t)r * Nn + k0 + c4);
      float4 a1 = *(const float4*)(Ab + (size_t)(r + 32) * Nn + k0 + c4);
      *(float4*)&As[r][c4]      = a0;
      *(float4*)&As[r + 32][c4] = a1;
    }
    {  // stage B panel: 16 K x 128 cols, interleaved as (k, k+1) pairs
      for (int idx = tid; idx < 256; idx += 128) {
        int p    = idx >> 5;           // kpair 0..7
        int j4   = (idx & 31) * 4;     // col 0..124 step 4
        int tseg = j4 >> 6;            // 0 or 1
        int c    = j4 & 63;
        const float* rp = Bb + (size_t)tseg * bStrideT +
                          (size_t)(k0 + 2 * p) * ldb + c;
        float4 va = *(const float4*)rp;         // row 2p
        float4 vb = *(const float4*)(rp + ldb); // row 2p+1
        float* dst = &Xs2[p * XPITCH + j4 * 2];
        *(float2*)(dst + 0) = make_float2(va.x, vb.x);
        *(float2*)(dst + 2) = make_float2(va.y, vb.y);
        *(float2*)(dst + 4) = make_float2(va.z, vb.z);
        *(float2*)(dst + 6) = make_float2(va.w, vb.w);
      }
    }
    if (k0 + 16 < Nn) {  // speculative L2 prefetch of next A panel
      __builtin_prefetch(Ab + (size_t)(tid >> 2) * Nn + k0 + 16, 0, 0);
    }
    __syncthreads();

#pragma unroll
    for (int kk = 0; kk < 16; kk += 4) {
      v2f a = *(const v2f*)&As[r0 + lm][kk + kh];
      const float* brow = &Xs2[((kk + kh) >> 1) * XPITCH];
#pragma unroll
      for (int ct = 0; ct < 8; ++ct) {
        v2f bb = *(const v2f*)(brow + (ct * 16 + lm) * 2);
        acc[ct] = wmma4(a, bb, acc[ct]);
      }
    }
  }

#pragma unroll
  for (int ct = 0; ct < 8; ++ct) {
#pragma unroll
    for (int i = 0; i < 8; ++i) {
      int row = mblk * 64 + r0 + i + hi;
      Ob[(size_t)row * COLS + ct * 16 + lm] = acc[ct][i];
    }
  }
}

// ---------------------------------------------------------------------------
// Kernel 2: fused 1x1 conv (K=192) + bias + BN + ReLU.
//   Gathers h rows from x / S1 / S2 (h never materialized).
//   Output time-major: ws_y[t][b][n][f].  64 rows per block (4 waves x 16).
// ---------------------------------------------------------------------------
__global__ __launch_bounds__(128) void conv_bn_relu_kernel(
    const float* __restrict__ x,    // [B][T][N][D] -> row m gives x + m*64
    const float* __restrict__ s1,   // [B][N][768]
    const float* __restrict__ s2,   // [B][N][768]
    const float* __restrict__ Wc,   // [F][192]
    const float* __restrict__ bc,   // [F]
    const float* __restrict__ gamma, const float* __restrict__ beta,
    const float* __restrict__ mean, const float* __restrict__ var,
    float* __restrict__ yout) {     // [T][B][N][F]
  const int tid  = threadIdx.x;
  const int lane = tid & 31;
  const int wave = tid >> 5;
  const int lm   = lane & 15;
  const int kh   = (lane >> 4) * 2;
  const int hi   = (lane >> 4) * 8;
  const int m0   = blockIdx.x * 64;

  __shared__ float hs[64][196];  // 64 rows x 192 (pitch 196, 16B aligned)

  // stage: 64 rows x 48 float4
  for (int idx = tid; idx < 64 * 48; idx += 128) {
    int lr  = idx / 48;
    int col = (idx % 48) * 4;
    int m   = m0 + lr;
    float4 v;
    if (col < 64) {
      v = *(const float4*)(x + (size_t)m * 64 + col);
    } else {
      int b = m / (Tn * Nn);
      int rem = m % (Tn * Nn);
      int t = rem / Nn;
      int n = rem % Nn;
      const float* src = (col < 128) ? s1 : s2;
      int c = (col < 128) ? (col - 64) : (col - 128);
      v = *(const float4*)(src + ((size_t)b * Nn + n) * COLS + t * 64 + c);
    }
    *(float4*)&hs[lr][col] = v;
  }
  __syncthreads();

  const int r0 = wave * 16;
  v8f acc[4] = {};
#pragma unroll 4
  for (int kk = 0; kk < CE; kk += 4) {
    v2f a = *(const v2f*)&hs[r0 + lm][kk + kh];
#pragma unroll
    for (int ct = 0; ct < 4; ++ct) {
      // B[k][f] = Wc[f][k]; (k,k+1) adjacent -> one global_load_b64
      v2f bb = *(const v2f*)&Wc[(size_t)(ct * 16 + lm) * CE + kk + kh];
      acc[ct] = wmma4(a, bb, acc[ct]);
    }
  }

#pragma unroll
  for (int ct = 0; ct < 4; ++ct) {
    int f = ct * 16 + lm;
    float scale = gamma[f] * rsqrtf(var[f] + EPS);
    float shift = beta[f] - mean[f] * scale;
    float bias  = bc[f];
#pragma unroll
    for (int i = 0; i < 8; ++i) {
      int m = m0 + r0 + i + hi;
      int b = m / (Tn * Nn);
      int rem = m % (Tn * Nn);
      int t = rem / Nn;
      int n = rem % Nn;
      float v = (acc[ct][i] + bias) * scale + shift;
      yout[(((size_t)t * Bn + b) * Nn + n) * Fn + f] = fmaxf(v, 0.0f);
    }
  }
}

// ---------------------------------------------------------------------------
// Kernel 3: one GRU timestep. 64 rows per block (4 waves x 16 rows).
//   cat = [x_t, h_prev] in LDS -> ru = sigmoid(cat@Wff+b) -> overwrite h-half
//   with r*h -> z = tanh(cat@Wzff+b) -> h = u*z + (1-u)*h_prev.
// ---------------------------------------------------------------------------
__global__ __launch_bounds__(128) void gru_step_kernel(
    const float* __restrict__ xs_t,   // [B*N][64] (time-major slice)
    const float* __restrict__ wff,    // [128][128] for this t
    const float* __restrict__ bff,    // [128]
    const float* __restrict__ wzff,   // [128][64]
    const float* __restrict__ bzff,   // [64]
    float* __restrict__ hstate,       // [B*N][64]
    float* __restrict__ out,          // [B][T][N][F]
    int t) {
  const int tid  = threadIdx.x;
  const int lane = tid & 31;
  const int wave = tid >> 5;
  const int lm   = lane & 15;
  const int kh   = (lane >> 4) * 2;
  const int hi   = (lane >> 4) * 8;
  const int gr0  = blockIdx.x * 64;

  __shared__ float cat[64][132];  // cols 0..63 = x_t, 64..127 = h (later r*h)

  for (int idx = tid; idx < 64 * 32; idx += 128) {
    int lr  = idx >> 5;
    int col = (idx & 31) * 4;
    const float* src = (col < 64) ? (xs_t + (size_t)(gr0 + lr) * Fn + col)
                                  : (hstate + (size_t)(gr0 + lr) * Fn + (col - 64));
    *(float4*)&cat[lr][col] = *(const float4*)src;
  }
  __syncthreads();

  const int r0 = wave * 16;

  // --- GEMM1: ru (128 cols, K=128) ---
  v8f ru[8] = {};
#pragma unroll 4
  for (int kk = 0; kk < 128; kk += 4) {
    v2f a = *(const v2f*)&cat[r0 + lm][kk + kh];
#pragma unroll
    for (int ct = 0; ct < 8; ++ct) {
      v2f bb;
      bb.x = wff[(size_t)(kk + kh) * 128 + ct * 16 + lm];
      bb.y = wff[(size_t)(kk + kh + 1) * 128 + ct * 16 + lm];
      ru[ct] = wmma4(a, bb, ru[ct]);
    }
  }
  // bias + sigmoid
#pragma unroll
  for (int ct = 0; ct < 8; ++ct) {
    float bias = bff[ct * 16 + lm];
#pragma unroll
    for (int i = 0; i < 8; ++i)
      ru[ct][i] = 1.0f / (1.0f + expf(-(ru[ct][i] + bias)));
  }

  // overwrite h-half of cat with r * h_prev (each wave owns its 16 rows)
#pragma unroll
  for (int ct = 0; ct < 4; ++ct) {
    int col = ct * 16 + lm;
#pragma unroll
    for (int i = 0; i < 8; ++i) {
      int row = r0 + i + hi;
      float hv = cat[row][64 + col];
      cat[row][64 + col] = ru[ct][i] * hv;
    }
  }
  __syncthreads();

  // --- GEMM2: z (64 cols, K=128) ---
  v8f zz[4] = {};
#pragma unroll 4
  for (int kk = 0; kk < 128; kk += 4) {
    v2f a = *(const v2f*)&cat[r0 + lm][kk + kh];
#pragma unroll
    for (int ct = 0; ct < 4; ++ct) {
      v2f bb;
      bb.x = wzff[(size_t)(kk + kh) * 64 + ct * 16 + lm];
      bb.y = wzff[(size_t)(kk + kh + 1) * 64 + ct * 16 + lm];
      zz[ct] = wmma4(a, bb, zz[ct]);
    }
  }

  // --- blend: hn = u*z + (1-u)*h_prev ; u tiles (4..7) align with z tiles ---
#pragma unroll
  for (int ct = 0; ct < 4; ++ct) {
    int col = ct * 16 + lm;
    float bz = bzff[col];
#pragma unroll
    for (int i = 0; i < 8; ++i) {
      int gr = gr0 + r0 + i + hi;
      float u  = ru[4 + ct][i];
      float zv = tanhf(zz[ct][i] + bz);
      float hp = hstate[(size_t)gr * Fn + col];
      float hn = u * zv + (1.0f - u) * hp;
      hstate[(size_t)gr * Fn + col] = hn;
      int b = gr / Nn, n = gr % Nn;
      out[(((size_t)b * Tn + t) * Nn + n) * Fn + col] = hn;
    }
  }
}

// ---------------------------------------------------------------------------
// Host launch
// ---------------------------------------------------------------------------
extern "C" void kernel_launch(void* const* d_in, const int* in_sizes, int n_in,
                              void* d_out, int out_size, void* d_ws, size_t ws_size,
                              hipStream_t stream) {
  const float* x      = (const float*)d_in[0];   // [B][T][N][D]
  const float* Agraph = (const float*)d_in[1];   // [B][N][N]
  const float* conv_w = (const float*)d_in[2];   // [F][192]
  const float* conv_b = (const float*)d_in[3];
  const float* gamma  = (const float*)d_in[4];
  const float* beta   = (const float*)d_in[5];
  const float* mean   = (const float*)d_in[6];
  const float* var    = (const float*)d_in[7];
  const float* wff    = (const float*)d_in[8];   // [T][128][128]
  const float* bff    = (const float*)d_in[9];   // [T][128]
  const float* wzff   = (const float*)d_in[10];  // [T][128][64]
  const float* bzff   = (const float*)d_in[11];  // [T][64]
  float* out = (float*)d_out;

  const size_t SZ = (size_t)Bn * Nn * COLS;  // 12,582,912 floats
  float* ws1 = (float*)d_ws;    // S1  [B][N][768]
  float* ws2 = ws1 + SZ;        // S2  [B][N][768]
  float* wsy = ws2 + SZ;        // y   [T][B][N][F] (same size)
  float* wsh = wsy + SZ;        // h   [B*N][F]

  // h0 = 0
  zero_kernel<<<512, 256, 0, stream>>>(wsh, (long long)Bn * Nn * Fn);

  dim3 ggrid(Nn / 64, COLS / 128, Bn);
  // S1 = A @ X0  (X0 rows come straight from x: ldb = 64)
  graph_gemm_kernel<<<ggrid, 128, 0, stream>>>(
      Agraph, x, ws1,
      (long long)Tn * Nn * Dn, (long long)Nn * Dn, Dn);
  // S2 = A @ S1  (workspace layout: ldb = 768)
  graph_gemm_kernel<<<ggrid, 128, 0, stream>>>(
      Agraph, ws1, ws2,
      (long long)Nn * COLS, (long long)64, COLS);

  // conv + BN + ReLU, time-major output
  conv_bn_relu_kernel<<<(Bn * Tn * Nn) / 64, 128, 0, stream>>>(
      x, ws1, ws2, conv_w, conv_b, gamma, beta, mean, var, wsy);

  // GRU scan over T (sequential dependence)
  for (int t = 0; t < Tn; ++t) {
    gru_step_kernel<<<(Bn * Nn) / 64, 128, 0, stream>>>(
        wsy + (size_t)t * Bn * Nn * Fn,
        wff + (size_t)t * 128 * 128,
        bff + (size_t)t * 128,
        wzff + (size_t)t * 128 * 64,
        bzff + (size_t)t * 64,
        wsh, out, t);
  }
}